// AnchorRadar_56977036148758
// MI455X (gfx1250) — compile-verified
//
#include <hip/hip_runtime.h>
#include <math.h>

typedef __attribute__((ext_vector_type(2))) float v2f;
typedef __attribute__((ext_vector_type(8))) float v8f;

#define D 64
#define INV_L1 (1.0f / 3.0f)   // 1/(L+1), L=2
#define EPS_NORM 1e-12f

// ---------------------------------------------------------------------------
// init: cur = [feat_a; feat_b];  acc (d_out block) = feat * 1/3
// ---------------------------------------------------------------------------
__global__ void init_kernel(const float* __restrict__ fa,
                            const float* __restrict__ fb,
                            size_t na64, size_t n64,
                            float* __restrict__ cur,
                            float* __restrict__ acc) {
    size_t i = (size_t)blockIdx.x * blockDim.x + threadIdx.x;
    if (i >= n64) return;
    float f = (i < na64) ? fa[i] : fb[i - na64];
    cur[i] = f;
    acc[i] = f * INV_L1;
}

// ---------------------------------------------------------------------------
// SpMM: next[row] += val * cur[col]   (16 lanes per edge, float4 per lane)
// Gathers hit L2 (node table <= 38.4MB << 192MB L2); scatter = f32 L2 atomics.
// ---------------------------------------------------------------------------
__global__ void spmm_kernel(const int* __restrict__ rows,
                            const int* __restrict__ cols,
                            const float* __restrict__ val,
                            const float* __restrict__ cur,
                            float* __restrict__ next,
                            int e2) {
    int t = blockIdx.x * blockDim.x + threadIdx.x;
    int e = t >> 4;
    if (e >= e2) return;
    int q = t & 15;
    int r = rows[e];
    int c = cols[e];
    float v = val[e];
    float4 m = ((const float4*)(cur + (size_t)c * D))[q];
    float* dst = next + (size_t)r * D + q * 4;
    unsafeAtomicAdd(dst + 0, m.x * v);
    unsafeAtomicAdd(dst + 1, m.y * v);
    unsafeAtomicAdd(dst + 2, m.z * v);
    unsafeAtomicAdd(dst + 3, m.w * v);
}

// ---------------------------------------------------------------------------
// Row-norm + accumulate. One wave per 16x64 f32 tile.
// V_WMMA_F32_16X16X4_F32 with B == A^T computes the Gram matrix; its diagonal
// is the per-row sum of squares. With the ISA's 16x4-A / 4x16-B f32 layouts,
// the A and B fragments for X and X^T are the IDENTICAL per-lane float2:
//   lane l<16 : (X[l][k0], X[l][k0+1]) ; lane 16+l : (X[l][k0+2], X[l][k0+3])
// so one 8-byte load per K-chunk feeds both operands. All 16 fragment loads
// are issued up front (one clause, one wait), then 16 chained f32 WMMAs cover
// K = 64. Diagonal broadcast uses v_readlane (constant lane per unrolled r),
// avoiding LDS bpermute traffic. acc += cur * (1/3)/max(sqrt(diag),1e-12).
// ---------------------------------------------------------------------------
__global__ void norm_acc_kernel(const float* __restrict__ cur,
                                float* __restrict__ acc,
                                int ntiles) {
    int lane = threadIdx.x & 31;
    int wave = threadIdx.x >> 5;
    int tile = blockIdx.x * 8 + wave;
    if (tile >= ntiles) return;              // wave-uniform: EXEC stays all-1s

    size_t row0 = (size_t)tile * 16;
    int sr   = lane & 15;                    // matrix row this lane feeds
    int cofs = (lane >> 4) << 1;             // 0 (K=0,1) or 2 (K=2,3)
    const float* base = cur + (row0 + sr) * D + cofs;

    // Stage all fragments first so the loads pipeline as one clause.
    float2 fv[16];
#pragma unroll
    for (int k = 0; k < 16; ++k)
        fv[k] = *(const float2*)(base + 4 * k);

    v8f g = {};                              // Gram accumulator (16x16 f32)
#pragma unroll
    for (int k = 0; k < 16; ++k) {
        v2f a;  a.x = fv[k].x;  a.y = fv[k].y;
        // D = A * A^T + C   (A and B fragments coincide)
        g = __builtin_amdgcn_wmma_f32_16x16x4_f32(
                /*neg_a=*/false, a, /*neg_b=*/false, a,
                /*c_mod=*/(short)0, g, /*reuse_a=*/false, /*reuse_b=*/false);
    }

    // Diagonal of D: row r<8 lives in VGPR r, lane r; row r>=8 in VGPR r-8,
    // lane r+16 (C/D layout: VGPR v -> M=v @ lanes 0-15, M=v+8 @ lanes 16-31).
    // Constant source lane per unrolled iteration -> v_readlane_b32 broadcast.
    float rn[16];
#pragma unroll
    for (int r = 0; r < 16; ++r) {
        int bits = __builtin_amdgcn_readlane(__float_as_int(g[r & 7]),
                                             (r < 8) ? r : (r + 16));
        float s = __int_as_float(bits);
        rn[r] = INV_L1 / fmaxf(sqrtf(s), EPS_NORM);
    }

    // acc += cur * rn  (each lane owns 2 columns -> coalesced 256B per row)
#pragma unroll
    for (int r = 0; r < 16; ++r) {
        size_t off = (row0 + r) * D + 2 * lane;
        float2 c = *(const float2*)(cur + off);
        float2* ap = (float2*)(acc + off);
        float2 av = *ap;
        av.x += c.x * rn[r];
        av.y += c.y * rn[r];
        *ap = av;
    }
}

// ---------------------------------------------------------------------------
// Host side
// ---------------------------------------------------------------------------
extern "C" void kernel_launch(void* const* d_in, const int* in_sizes, int n_in,
                              void* d_out, int out_size, void* d_ws, size_t ws_size,
                              hipStream_t stream) {
    const float* users   = (const float*)d_in[0];   // 100000 x 64
    const float* bundles = (const float*)d_in[1];   //  20000 x 64
    const float* items   = (const float*)d_in[2];   //  50000 x 64

    struct Graph {
        const int*   idx;   // [2, E2] : rows then cols
        const float* val;   // [E2]
        int          e2;
        const float* fa;
        const float* fb;
        int          na, nb;
        size_t       outrow; // row offset of this graph's block in d_out
    };
    Graph gs[3] = {
        { (const int*)d_in[3], (const float*)d_in[4], in_sizes[4],
          users,   bundles, 100000, 20000, 0      },   // UB -> rows [0,120000)
        { (const int*)d_in[5], (const float*)d_in[6], in_sizes[6],
          users,   items,   100000, 50000, 120000 },   // UI -> rows [120000,270000)
        { (const int*)d_in[7], (const float*)d_in[8], in_sizes[8],
          bundles, items,    20000, 50000, 270000 },   // BI -> rows [270000,340000)
    };

    // ping-pong node buffers in workspace (max n = 150000 rows)
    const size_t maxn64 = (size_t)150000 * D;
    float* buf0 = (float*)d_ws;
    float* buf1 = buf0 + maxn64;
    float* out  = (float*)d_out;

    for (int gi = 0; gi < 3; ++gi) {
        const Graph& g = gs[gi];
        const int n      = g.na + g.nb;
        const size_t n64 = (size_t)n * D;
        float* cur = buf0;
        float* nxt = buf1;
        float* acc = out + g.outrow * D;

        init_kernel<<<(unsigned)((n64 + 255) / 256), 256, 0, stream>>>(
            g.fa, g.fb, (size_t)g.na * D, n64, cur, acc);

        const int* rows = g.idx;
        const int* cols = g.idx + g.e2;
        const int ntiles = n / 16;   // n is a multiple of 16 for all graphs

        for (int layer = 0; layer < 2; ++layer) {
            hipMemsetAsync(nxt, 0, n64 * sizeof(float), stream);

            int threads = g.e2 * 16;
            spmm_kernel<<<(threads + 255) / 256, 256, 0, stream>>>(
                rows, cols, g.val, cur, nxt, g.e2);

            norm_acc_kernel<<<(ntiles + 7) / 8, 256, 0, stream>>>(nxt, acc, ntiles);

            float* t = cur; cur = nxt; nxt = t;
        }
    }
}